// DCGRUCell_50302656971158
// MI455X (gfx1250) — compile-verified
//
#include <hip/hip_runtime.h>
#include <hip/hip_bf16.h>

typedef __attribute__((ext_vector_type(2))) float v2f;
typedef __attribute__((ext_vector_type(8))) float v8f;

#define NN 4096   // num nodes
#define BB 32     // batch
#define UU 64     // units
#define DD 64     // input dim
#define SS 128    // D + U
#define MMAT 3    // num diffusion matrices
#define CC 4096   // columns of X matrices = S*B

// ---- CDNA5 async global->LDS copy (ASYNCcnt-tracked, no VGPR round trip) ----
// LDS flat addresses keep the byte offset in addr[31:0] (ISA aperture rules),
// so truncating the generic pointer yields the LDS address operand.
__device__ __forceinline__ unsigned lds_addr32(const void* p) {
    return (unsigned)(unsigned long long)p;
}
__device__ __forceinline__ void async_ld_b128(unsigned ldsoff, const float* gptr) {
    asm volatile("global_load_async_to_lds_b128 %0, %1, off"
                 :: "v"(ldsoff), "v"(gptr)
                 : "memory");
}
__device__ __forceinline__ void wait_async0() {
    asm volatile("s_wait_asynccnt 0x0" ::: "memory");
}

// ---------------- build X0 = concat(inputs, hx) laid out (N, S*B) -------------
__global__ __launch_bounds__(256) void k_build_x0(float* __restrict__ X0,
        const float* __restrict__ inp, const float* __restrict__ hx)
{
    int idx = blockIdx.x * 256 + threadIdx.x;      // over N*CC = 16.7M
    int n = idx >> 12;
    int c = idx & 4095;
    int s = c >> 5;
    int b = c & 31;
    float v;
    if (s < DD) v = inp[(size_t)b * (NN * DD) + (size_t)n * DD + s];
    else        v = hx [(size_t)b * (NN * UU) + (size_t)n * UU + (s - DD)];
    X0[idx] = v;
}

// ---------------- Y = alpha*(A@X) + beta*Z  (all 4096x4096 fp32) --------------
// Block tile: 128(M) x 128(N), 8 waves; wave (wm 0..3, wn 0..1) owns 32 x 64
// (2 row strips x 4 col tiles = 8 accumulators). K staged 32-wide through
// double-buffered LDS via async global->LDS copies; fp32 WMMA 16x16x4 core.
__global__ __launch_bounds__(256) void k_gemm(float* __restrict__ Y,
        const float* __restrict__ A, const float* __restrict__ X,
        const float* __restrict__ Z, float alpha, float beta)
{
    __shared__ float As[2][128][36];   // 144B row stride (16B aligned)
    __shared__ float Xs[2][32][136];   // 544B row stride (16B aligned)

    const int tid   = threadIdx.x;
    const int lane  = tid & 31;
    const int wid   = tid >> 5;
    const int wm    = wid >> 1;        // 0..3 -> 32-row strip pair
    const int wn    = wid & 1;         // 0..1 -> 64-col half
    const int lmod  = lane & 15;
    const int lhalf = lane >> 4;
    const int m0    = blockIdx.y * 128;
    const int n0    = blockIdx.x * 128;

    // per-thread async staging of one K-slab into LDS buffer `buf`
    auto stage = [&](int buf, int kb) {
#pragma unroll
        for (int i = 0; i < 4; ++i) {          // A tile 128x32
            int v  = tid + i * 256;
            int r  = v >> 3;
            int c4 = (v & 7) << 2;
            async_ld_b128(lds_addr32(&As[buf][r][c4]),
                          &A[(size_t)(m0 + r) * NN + kb + c4]);
        }
#pragma unroll
        for (int i = 0; i < 4; ++i) {          // X tile 32x128
            int v  = tid + i * 256;
            int r  = v >> 5;
            int c4 = (v & 31) << 2;
            async_ld_b128(lds_addr32(&Xs[buf][r][c4]),
                          &X[(size_t)(kb + r) * CC + n0 + c4]);
        }
    };

    v8f acc[2][4];
#pragma unroll
    for (int rr = 0; rr < 2; ++rr)
#pragma unroll
        for (int t = 0; t < 4; ++t) acc[rr][t] = v8f{};

    stage(0, 0);
    for (int kb = 0; kb < NN; kb += 32) {
        const int ibuf = (kb >> 5) & 1;
        wait_async0();                 // my copies into ibuf have landed in LDS
        __syncthreads();               // everyone's landed; prev buffer reads done
        if (kb + 32 < NN) stage(ibuf ^ 1, kb + 32);   // prefetch next slab

        const int arow = wm * 32 + lmod;
#pragma unroll
        for (int kk = 0; kk < 32; kk += 4) {
            int kv = kk + 2 * lhalf;              // lane-half selects K {0,1} vs {2,3}
            v2f a0, a1;
            a0.x = As[ibuf][arow][kv];
            a0.y = As[ibuf][arow][kv + 1];
            a1.x = As[ibuf][arow + 16][kv];
            a1.y = As[ibuf][arow + 16][kv + 1];
#pragma unroll
            for (int t = 0; t < 4; ++t) {
                int bcol = wn * 64 + t * 16 + lmod;
                v2f bv;
                bv.x = Xs[ibuf][kv][bcol];
                bv.y = Xs[ibuf][kv + 1][bcol];
                acc[0][t] = __builtin_amdgcn_wmma_f32_16x16x4_f32(
                    false, a0, false, bv, (short)0, acc[0][t], false, false);
                acc[1][t] = __builtin_amdgcn_wmma_f32_16x16x4_f32(
                    false, a1, false, bv, (short)0, acc[1][t], false, false);
            }
        }
        __syncthreads();               // all reads of ibuf done before it is reused
    }

    // epilogue: C/D layout -> VGPR i = row (i + 8*lane_half), col = lane&15
#pragma unroll
    for (int rr = 0; rr < 2; ++rr) {
        int rowbase = m0 + wm * 32 + rr * 16;
#pragma unroll
        for (int t = 0; t < 4; ++t) {
            int col = n0 + wn * 64 + t * 16 + lmod;
#pragma unroll
            for (int i = 0; i < 8; ++i) {
                int row = rowbase + i + 8 * lhalf;
                float y = alpha * acc[rr][t][i];
                if (beta != 0.0f) y += beta * Z[(size_t)row * CC + col];
                Y[(size_t)row * CC + col] = y;
            }
        }
    }
}

// ------- projection: out(B*N, OD) = act( gather(X0,X1,X2) @ W + bias ) --------
// Row R = b*N + n; feature k = m*128 + s maps to X_m[n, s*32+b], W row = s*3+m.
// One wave per 16x16 output tile, K = 384 in steps of 4 (fp32 WMMA).
template<int OD, int ACT>
__global__ __launch_bounds__(256) void k_proj(float* __restrict__ out,
        const float* __restrict__ X0, const float* __restrict__ X1,
        const float* __restrict__ X2,
        const float* __restrict__ W, const float* __restrict__ bias)
{
    const int lane   = threadIdx.x & 31;
    const int wid    = threadIdx.x >> 5;
    const int waveId = blockIdx.x * 8 + wid;
    const int CT     = OD / 16;
    const int ct     = waveId % CT;
    const int rt     = waveId / CT;
    const int R0     = rt * 16;
    const int b      = R0 >> 12;      // batch (tile never crosses batch boundary)
    const int n0     = R0 & 4095;
    const int lmod   = lane & 15;
    const int lhalf  = lane >> 4;
    const int nrow   = n0 + lmod;
    const int o      = ct * 16 + lmod;

    v8f acc = v8f{};
    for (int k0 = 0; k0 < SS * MMAT; k0 += 4) {
        int kb = k0 + 2 * lhalf;
        v2f a, w;
#pragma unroll
        for (int j = 0; j < 2; ++j) {
            int k = kb + j;
            int m = k >> 7;          // which diffusion matrix
            int s = k & 127;         // which feature
            const float* Xp = (m == 0) ? X0 : ((m == 1) ? X1 : X2);
            a[j] = Xp[(size_t)nrow * CC + s * BB + b];
            w[j] = W[(size_t)(s * MMAT + m) * OD + o];
        }
        acc = __builtin_amdgcn_wmma_f32_16x16x4_f32(
            false, a, false, w, (short)0, acc, false, false);
    }

    float bv = bias[o];
#pragma unroll
    for (int i = 0; i < 8; ++i) {
        int r = R0 + i + 8 * lhalf;
        float y = acc[i] + bv;
        y = (ACT == 0) ? (1.0f / (1.0f + __expf(-y))) : tanhf(y);
        out[(size_t)r * OD + (ct * 16 + lmod)] = y;
    }
}

// -------- overwrite state half of X0 with r*h (in place, X0 dead by now) ------
__global__ __launch_bounds__(256) void k_update_state(float* __restrict__ X0,
        const float* __restrict__ ru, const float* __restrict__ hx)
{
    int idx = blockIdx.x * 256 + threadIdx.x;     // over B*N*U
    int b = idx >> 18;
    int n = (idx >> 6) & 4095;
    int j = idx & 63;
    float r = ru[(size_t)(b * NN + n) * (2 * UU) + j];
    float h = hx[(size_t)b * (NN * UU) + (size_t)n * UU + j];
    X0[(size_t)n * CC + (DD + j) * BB + b] = r * h;
}

// -------- new_h = u*h + (1-u)*c ----------------------------------------------
__global__ __launch_bounds__(256) void k_combine(float* __restrict__ out,
        const float* __restrict__ ru, const float* __restrict__ cbuf,
        const float* __restrict__ hx)
{
    int idx = blockIdx.x * 256 + threadIdx.x;     // over B*N*U; matches hx/out layout
    int b = idx >> 18;
    int n = (idx >> 6) & 4095;
    int j = idx & 63;
    float u = ru[(size_t)(b * NN + n) * (2 * UU) + UU + j];
    float c = cbuf[(size_t)(b * NN + n) * UU + j];
    float h = hx[idx];
    out[idx] = u * h + (1.0f - u) * c;
}

extern "C" void kernel_launch(void* const* d_in, const int* in_sizes, int n_in,
                              void* d_out, int out_size, void* d_ws, size_t ws_size,
                              hipStream_t stream) {
    const float* inp  = (const float*)d_in[0];
    const float* hx   = (const float*)d_in[1];
    const float* sup  = (const float*)d_in[2];
    const float* W_ru = (const float*)d_in[3];
    const float* b_ru = (const float*)d_in[4];
    const float* W_c  = (const float*)d_in[5];
    const float* b_c  = (const float*)d_in[6];
    float* out = (float*)d_out;

    float* ws = (float*)d_ws;
    const size_t XSZ = (size_t)NN * CC;            // 16.7M floats each
    float* X0 = ws;
    float* X1 = X0 + XSZ;
    float* X2 = X1 + XSZ;
    float* ru = X2 + XSZ;                          // (B*N, 128)
    float* cb = ru + (size_t)BB * NN * 2 * UU;     // (B*N, 64)

    dim3 gemmGrid(CC / 128, NN / 128);             // (32, 32)
    const int elemBlocks = (BB * NN * UU) / 256;   // 32768

    // ---- gate path ----
    k_build_x0<<<(NN * CC) / 256, 256, 0, stream>>>(X0, inp, hx);
    k_gemm<<<gemmGrid, 256, 0, stream>>>(X1, sup, X0, X0, 1.0f, 0.0f);
    k_gemm<<<gemmGrid, 256, 0, stream>>>(X2, sup, X1, X0, 2.0f, -1.0f);
    k_proj<128, 0><<<(BB * NN / 16) * (128 / 16) / 8, 256, 0, stream>>>(
        ru, X0, X1, X2, W_ru, b_ru);

    // ---- candidate path (reuse X buffers) ----
    k_update_state<<<elemBlocks, 256, 0, stream>>>(X0, ru, hx);
    k_gemm<<<gemmGrid, 256, 0, stream>>>(X1, sup, X0, X0, 1.0f, 0.0f);
    k_gemm<<<gemmGrid, 256, 0, stream>>>(X2, sup, X1, X0, 2.0f, -1.0f);
    k_proj<64, 1><<<(BB * NN / 16) * (64 / 16) / 8, 256, 0, stream>>>(
        cb, X0, X1, X2, W_c, b_c);

    // ---- GRU combine ----
    k_combine<<<elemBlocks, 256, 0, stream>>>(out, ru, cb, hx);
}